// ContextWindow_14869176779353
// MI455X (gfx1250) — compile-verified
//
#include <hip/hip_runtime.h>
#include <cstdint>

#define B_      32
#define T_      2000
#define C_      80
#define CTX     11
#define LEFTP   5
#define TT      64                 // time-steps per block
#define ROWS    (TT + 10)          // 74 rows incl. halo
#define RSTRIDE 84                 // padded LDS row stride (floats), 336B = 16B-aligned
#define NTHREADS 256

__global__ __launch_bounds__(NTHREADS)
void ContextWindow_kernel(const float* __restrict__ x,
                          const float* __restrict__ lengths,
                          float* __restrict__ out) {
    __shared__ float tile[ROWS * RSTRIDE];

    const int tid     = threadIdx.x;
    const int tile_id = blockIdx.x;
    const int b       = blockIdx.y;
    const int t0      = tile_id * TT;

    // ---- Phase 1: zero LDS tile (so out-of-range halo rows read as 0) ----
    {
        float4* t4 = (float4*)tile;
        const float4 z = make_float4(0.f, 0.f, 0.f, 0.f);
        const int n4 = (ROWS * RSTRIDE) / 4;           // 1554
        for (int i = tid; i < n4; i += NTHREADS) t4[i] = z;
    }
    __syncthreads();

    // ---- Phase 2: async copy rows [t0-5, t0+TT+5) clipped to [0,T) into LDS ----
    {
        // low 32 bits of the flat pointer to a __shared__ object == LDS byte offset
        const uint32_t lds_base = (uint32_t)(uint64_t)(uintptr_t)tile;
        const int nchunks = ROWS * (C_ / 4);           // 74 * 20 = 1480 b128 chunks
        for (int i = tid; i < nchunks; i += NTHREADS) {
            const int row  = i / 20;
            const int q    = i - row * 20;             // 16-byte chunk within row
            const int grow = t0 - LEFTP + row;
            if (grow >= 0 && grow < T_) {
                const uint32_t lds_addr =
                    lds_base + (uint32_t)(row * RSTRIDE + q * 4) * 4u;
                const float* gsrc = x + ((size_t)b * T_ + (size_t)grow) * C_ + q * 4;
                asm volatile("global_load_async_to_lds_b128 %0, %1, off"
                             :: "v"(lds_addr), "v"(gsrc)
                             : "memory");
            }
        }
        asm volatile("s_wait_asynccnt 0x0" ::: "memory");
    }
    __syncthreads();

    // ---- Phase 3: emit 11 shifted copies per (t, 4-channel group) ----
    const int len_abs = (int)rintf((float)T_ * lengths[b]);

    const int npairs = TT * (C_ / 4);                  // 1280 pairs / block
    for (int p = tid; p < npairs; p += NTHREADS) {
        const int tt = p / 20;
        const int g  = p - tt * 20;
        const int t  = t0 + tt;
        if (t >= T_) continue;
        const float m  = (t < len_abs) ? 1.0f : 0.0f;
        const int   c0 = g * 4;

        float r[CTX][4];
#pragma unroll
        for (int i = 0; i < CTX; ++i) {
            const float4 v = *(const float4*)&tile[(tt + i) * RSTRIDE + c0];
            r[i][0] = v.x; r[i][1] = v.y; r[i][2] = v.z; r[i][3] = v.w;
        }

        // out[b, t, c*11 + i]; thread owns 44 consecutive floats at c0*11 (16B aligned)
        float* op = out + ((size_t)b * T_ + (size_t)t) * (size_t)(C_ * CTX)
                        + (size_t)c0 * CTX;
#pragma unroll
        for (int s = 0; s < CTX; ++s) {
            float4 o;
            o.x = m * r[(4 * s + 0) % CTX][(4 * s + 0) / CTX];
            o.y = m * r[(4 * s + 1) % CTX][(4 * s + 1) / CTX];
            o.z = m * r[(4 * s + 2) % CTX][(4 * s + 2) / CTX];
            o.w = m * r[(4 * s + 3) % CTX][(4 * s + 3) / CTX];
            *(float4*)(op + 4 * s) = o;
        }
    }
}

extern "C" void kernel_launch(void* const* d_in, const int* in_sizes, int n_in,
                              void* d_out, int out_size, void* d_ws, size_t ws_size,
                              hipStream_t stream) {
    (void)in_sizes; (void)n_in; (void)ws_size; (void)d_ws; (void)out_size;
    const float* x       = (const float*)d_in[0];
    const float* lengths = (const float*)d_in[1];
    float*       out     = (float*)d_out;

    dim3 grid((T_ + TT - 1) / TT, B_);   // 32 x 32 = 1024 blocks
    ContextWindow_kernel<<<grid, NTHREADS, 0, stream>>>(x, lengths, out);
}